// MXQuantizer_87273735454890
// MI455X (gfx1250) — compile-verified
//
#include <hip/hip_runtime.h>
#include <cstdint>
#include <cstddef>

// ---------------------------------------------------------------------------
// MX (mxfp8-e4m3, block=32) fake-quantization of fp32, tuned for MI455X:
//  * streaming roofline: 512MB traffic -> ~22us @ 23.3 TB/s.
//  * CDNA5 async path: global_load_async_to_lds_b128 + s_wait_asynccnt,
//    4-buffer rotation, prefetch depth 3 (1.5KB in flight per wave) --
//    the buffer being refilled was consumed a full iteration earlier, so
//    the async-LDS-write vs ds_load WAR window is closed architecturally.
//  * block-of-32 max|x|: lane owns float4, 8 lanes per MX block,
//    3x dpp8 (xor 1/2/4) + v_max -- pure VALU, no LDS pipe contention.
//  * VALU-lean element path: v_frexp_exp_i32_f32 for floor(log2),
//    v_ldexp_f32 with REGISTER exponent for all power-of-2 scaling.
//  * non-temporal B128 stores (512MB stream >> 192MB L2, zero reuse).
// ---------------------------------------------------------------------------

typedef float v4f __attribute__((ext_vector_type(4)));

constexpr int THREADS = 256;   // 8 waves per workgroup
constexpr int NITER   = 8;     // float4 chunks per thread in the main kernel
constexpr int BUFS    = 4;     // LDS staging buffers (rotation)

#define ASYNC_LOAD_B128(ldsAddr, gAddr)                                        \
  asm volatile("global_load_async_to_lds_b128 %0, %1, off"                     \
               :: "v"(ldsAddr), "v"(gAddr) : "memory")

#define WAIT_ASYNC(n) asm volatile("s_wait_asynccnt " #n ::: "memory")

// frexp-style exponent: v = m * 2^e, m in [0.5,1) -> returns e.
// e == floor(log2(v)) + 1 for v > 0 (denormal-correct in HW).
// Returns 0 for +-0 / Inf / NaN (hardware semantics) -- callers rely on this.
__device__ __forceinline__ int frexp_e(float v) {
#if __has_builtin(__builtin_amdgcn_frexp_expf)
  return __builtin_amdgcn_frexp_expf(v);
#else
  unsigned b = __float_as_uint(v) & 0x7FFFFFFFu;
  if (b == 0 || b >= 0x7F800000u) return 0;
  int e = (int)(b >> 23);
  if (e) return e - 126;
  return (31 - __clz((int)(b & 0x7FFFFFu))) - 148;
#endif
}

// max across the 8-lane group that owns one 32-element MX block.
// dpp8 selectors implement lane^1, lane^2, lane^4 within each group of 8.
__device__ __forceinline__ float blockmax8(float v) {
  int b;
  b = __builtin_amdgcn_mov_dpp8(__float_as_int(v), 0xDE54C1); // xor 1
  v = fmaxf(v, __int_as_float(b));
  b = __builtin_amdgcn_mov_dpp8(__float_as_int(v), 0xB3E21A); // xor 2
  v = fmaxf(v, __int_as_float(b));
  b = __builtin_amdgcn_mov_dpp8(__float_as_int(v), 0x688FAC); // xor 4
  v = fmaxf(v, __int_as_float(b));
  return v;
}

// e4m3 quantization of a = x/scale, then re-scale.
// Reference: pe = max(floor(log2|a|), -6); q = a*2^(3-pe);
// r = sign(q)*floor(|q|+0.5); o = r*2^(pe-3); clip +-448; NaN restored.
// With fe = frexp_e(|a|) = floor(log2|a|)+1:  u = 3-pe = min(4-fe, 9).
// a==0 -> fe=0 -> u=4 -> q=0 -> o=0 (matches reference output 0).
__device__ __forceinline__ float quant_elem(float a, float scale) {
  int fe = frexp_e(fabsf(a));
  int u  = 4 - fe;
  u = u > 9 ? 9 : u;                                // private-exp clamp (pe >= -6)
  float q = ldexpf(a, u);                           // v_ldexp_f32, reg exponent
  float r = copysignf(floorf(fabsf(q) + 0.5f), q);  // ties away from zero
  float o = ldexpf(r, -u);
  o = fminf(fmaxf(o, -448.0f), 448.0f);             // saturate (IEEE *Num: drops NaN)
  if (a != a) o = __uint_as_float(0x7FC00000u);     // restore NaN
  return o * scale;
}

__device__ __forceinline__ v4f process4(v4f x) {
  float m = fmaxf(fmaxf(fabsf(x.x), fabsf(x.y)), fmaxf(fabsf(x.z), fabsf(x.w)));
  float bmax = blockmax8(m);

  // shared_exp = clamp(floor(log2(bmax)) - 8, >= -127); Inf/NaN bmax -> NaN scale.
  // floor(log2(bmax)) - 8 = frexp_e(bmax) - 9.
  int se = frexp_e(bmax) - 9;
  if (se < -127) se = -127;
  if (bmax == 0.0f) se = -127;                      // ref: -126-8 clamped to -127
  float scale  = ldexpf(1.0f,  se);                 // 2^-127 denorm OK (denorms on)
  float iscale = ldexpf(1.0f, -se);                 // exact: x/scale == x*iscale
  unsigned bb = __float_as_uint(bmax);
  if ((bb & 0x7F800000u) == 0x7F800000u) {          // Inf/NaN block max
    scale = iscale = __uint_as_float(0x7FC00000u);
  }

  v4f r;
  r.x = quant_elem(x.x * iscale, scale);
  r.y = quant_elem(x.y * iscale, scale);
  r.z = quant_elem(x.z * iscale, scale);
  r.w = quant_elem(x.w * iscale, scale);
  return r;
}

// Main kernel: async global->LDS streaming pipeline, 4-buffer rotation,
// prefetch depth 3. grid*THREADS*NITER == number of float4 chunks covered.
__global__ __launch_bounds__(THREADS) void mxq_main(const float* __restrict__ in,
                                                    float* __restrict__ out) {
  __shared__ __align__(16) v4f lbuf[BUFS][THREADS];

  const unsigned tid    = threadIdx.x;
  const unsigned stride = blockDim.x * gridDim.x;
  const unsigned idx0   = blockIdx.x * blockDim.x + tid;

  // LDS byte address of this lane's slot (generic-ptr low 32 bits == LDS offset)
  const unsigned lds_t = (unsigned)(uintptr_t)&lbuf[0][0] + tid * 16u;

  auto issue = [&](int chunkIt) {
    unsigned c   = idx0 + (unsigned)chunkIt * stride;
    unsigned lds = lds_t + ((unsigned)chunkIt & (BUFS - 1)) * (THREADS * 16u);
    uint64_t g   = (uint64_t)(uintptr_t)(in + (size_t)c * 4);
    ASYNC_LOAD_B128(lds, g);
  };

  // prologue: prefetch chunks 0..2 (depth 3)
  issue(0);
  issue(1);
  issue(2);

#pragma unroll
  for (int it = 0; it < NITER; ++it) {
    // In-order async completion: after issuing it+3, allowing 3 outstanding
    // guarantees chunk `it` has landed in LDS. Drain tail: 2 -> 1 -> 0.
    if (it + 3 < NITER) { issue(it + 3); WAIT_ASYNC(3); }
    else if (it + 2 < NITER) { WAIT_ASYNC(2); }
    else if (it + 1 < NITER) { WAIT_ASYNC(1); }
    else { WAIT_ASYNC(0); }

    v4f x = lbuf[it & (BUFS - 1)][tid]; // ds_load_b128
    v4f r = process4(x);

    unsigned chunk = idx0 + (unsigned)it * stride;
    __builtin_nontemporal_store(r, (v4f*)(out + (size_t)chunk * 4)); // NT B128 store
  }
}

// Tail kernel for chunk counts not divisible by THREADS*NITER (unused for 8192^2).
// n4 is a multiple of 8 (rows are multiples of 32 floats) so 8-lane groups stay whole.
__global__ __launch_bounds__(THREADS) void mxq_tail(const float* __restrict__ in,
                                                    float* __restrict__ out,
                                                    unsigned start4, unsigned n4) {
  unsigned c = start4 + blockIdx.x * blockDim.x + threadIdx.x;
  if (c >= n4) return;
  v4f x = __builtin_nontemporal_load((const v4f*)(in + (size_t)c * 4));
  v4f r = process4(x);
  __builtin_nontemporal_store(r, (v4f*)(out + (size_t)c * 4));
}

extern "C" void kernel_launch(void* const* d_in, const int* in_sizes, int n_in,
                              void* d_out, int out_size, void* d_ws, size_t ws_size,
                              hipStream_t stream) {
  (void)n_in; (void)d_ws; (void)ws_size; (void)out_size;
  const float* in = (const float*)d_in[0];
  float* out = (float*)d_out;

  const size_t n  = (size_t)in_sizes[0];   // 8192*8192 = 67,108,864 floats
  const unsigned n4 = (unsigned)(n / 4);   // 16,777,216 float4 chunks

  const unsigned per_block = THREADS * NITER;      // 2048 chunks per block
  const unsigned blocks    = n4 / per_block;       // 8192 blocks for 8192^2
  const unsigned covered   = blocks * per_block;

  if (blocks)
    mxq_main<<<blocks, THREADS, 0, stream>>>(in, out);

  const unsigned rem = n4 - covered;
  if (rem) {
    unsigned tb = (rem + THREADS - 1) / THREADS;
    mxq_tail<<<tb, THREADS, 0, stream>>>(in, out, covered, n4);
  }
}